// Finding_Correspondence_66623532696122
// MI455X (gfx1250) — compile-verified
//
#include <hip/hip_runtime.h>

// Finding_Correspondence for MI455X (gfx1250, wave32).
// Pure memory-bound gather/scatter: no matmul content -> WMMA inapplicable.
// Perf levers used:
//   * 4 elements/thread: 4 independent random-gather chains per lane ->
//     4x memory-level parallelism per wave (the real limiter for gathers).
//   * 128-bit non-temporal loads/stores (th:TH_*_NT) on all streaming data,
//     reserving the 192MB L2 for the randomly-gathered tables
//     (facets_corr + pos: 6MB hot per batch, L2-resident, ~4 hits/line).
//   * global_prefetch_b8 on the dependent gather addresses.
//   * deterministic atomicMax for both duplicate-index scatters.

typedef int   v4i __attribute__((ext_vector_type(4)));
typedef float v4f __attribute__((ext_vector_type(4)));

static constexpr int kB = 64;
static constexpr int kN = 200000;           // divisible by 4
static constexpr int kN4 = kN / 4;          // 50000
static constexpr int kF = 12 * 256 * 256;   // 786432
static constexpr int BLK = 256;             // 8 wave32 waves per block

// ---------------------------------------------------------------------------
// Fill an int region with -1 using 128-bit non-temporal stores (grid-stride).
// ---------------------------------------------------------------------------
__global__ void fc_fill_neg1_i4(v4i* __restrict__ p, long long n4) {
  long long i      = (long long)blockIdx.x * blockDim.x + threadIdx.x;
  long long stride = (long long)gridDim.x * blockDim.x;
  v4i v = {-1, -1, -1, -1};
  for (; i < n4; i += stride) {
    __builtin_nontemporal_store(v, &p[i]);
  }
}

// ---------------------------------------------------------------------------
// Initialize the gt_matches1 plane (out[b][1][:]) to -1.0f, 128-bit NT stores.
// ---------------------------------------------------------------------------
__global__ void fc_init_gt1(float* __restrict__ out) {
  int b = blockIdx.y;
  long long base = (long long)b * 3 * kN + kN;   // [b][1][0]
  int i4 = blockIdx.x * blockDim.x + threadIdx.x;
  if (i4 < kN4) {
    v4f v = {-1.0f, -1.0f, -1.0f, -1.0f};
    __builtin_nontemporal_store(v, (v4f*)(out + base) + i4);
  }
}

// ---------------------------------------------------------------------------
// pos[b][idx1[b][i]] = i, 4 elements per thread. Duplicates resolved
// deterministically by atomicMax (largest i wins; JAX scatter order is
// unspecified, so any deterministic winner is valid).
// ---------------------------------------------------------------------------
__global__ void fc_scatter_pos(const int* __restrict__ idx1,
                               int* __restrict__ pos) {
  int b  = blockIdx.y;
  int t4 = blockIdx.x * blockDim.x + threadIdx.x;
  if (t4 >= kN4) return;
  long long baseN = (long long)b * kN;
  long long baseF = (long long)b * kF;

  v4i f = __builtin_nontemporal_load((const v4i*)(idx1 + baseN) + t4);
  int i0 = t4 * 4;
#pragma unroll
  for (int k = 0; k < 4; ++k) {
    int fk = f[k];
    if ((unsigned)fk < (unsigned)kF) {
      atomicMax(&pos[baseF + fk], i0 + k);
    }
  }
}

// ---------------------------------------------------------------------------
// Main gather + outputs + gt_matches1 scatter, 4 elements per thread:
//   f0   = idx0[b][i]                        (128-bit NT streaming load)
//   c    = facets_corr[b][f0]                (random gather, prefetched)
//   m0   = c >= 0 ? pos[b][min(c,F-1)] : -1  (random gather)
//   out[b][0][i] = (float)m0                          (128-bit NT store)
//   out[b][2][i] = m0 >= 0 ? scores[b][i] : 0         (128-bit NT store)
//   if m0 >= 0: out[b][1][m0] = max(prev, (float)i)   (deterministic atomic)
// atomicMax on int bits is exact: scattered values are non-negative floats
// (monotone in int bit pattern) and the -1.0f init is negative as an int.
// ---------------------------------------------------------------------------
__global__ void fc_main(const int* __restrict__ corr,
                        const int* __restrict__ idx0,
                        const int* __restrict__ pos,
                        const float* __restrict__ scores,
                        float* __restrict__ out) {
  int b  = blockIdx.y;
  int t4 = blockIdx.x * blockDim.x + threadIdx.x;
  if (t4 >= kN4) return;

  long long baseN = (long long)b * kN;
  long long baseF = (long long)b * kF;

  v4i f0 = __builtin_nontemporal_load((const v4i*)(idx0 + baseN) + t4);
  // Kick all 4 independent dependent-gather chains before touching them.
#pragma unroll
  for (int k = 0; k < 4; ++k) {
    __builtin_prefetch(&corr[baseF + f0[k]], 0, 3);
  }
  v4f s = __builtin_nontemporal_load((const v4f*)(scores + baseN) + t4);

  // First-level gathers (4 independent chains in flight).
  int c0 = corr[baseF + f0[0]];
  int c1 = corr[baseF + f0[1]];
  int c2 = corr[baseF + f0[2]];
  int c3 = corr[baseF + f0[3]];
  v4i c = {c0, c1, c2, c3};

  // Second-level gathers.
  v4i m0;
#pragma unroll
  for (int k = 0; k < 4; ++k) {
    int ck = c[k];
    if (ck >= 0) {
      int cc = (ck < kF - 1) ? ck : (kF - 1);   // reference clips to [0, F-1]
      m0[k] = pos[baseF + cc];
    } else {
      m0[k] = -1;
    }
  }

  v4f m0f, sm;
#pragma unroll
  for (int k = 0; k < 4; ++k) {
    m0f[k] = (float)m0[k];
    sm[k]  = (m0[k] >= 0) ? s[k] : 0.0f;
  }

  float* ob = out + (long long)b * 3 * kN;
  __builtin_nontemporal_store(m0f, (v4f*)ob + t4);                // [b][0][i..i+3]
  __builtin_nontemporal_store(sm, (v4f*)(ob + 2LL * kN) + t4);    // [b][2][i..i+3]

  int i0 = t4 * 4;
  int* gt1 = (int*)(ob + (long long)kN);
#pragma unroll
  for (int k = 0; k < 4; ++k) {
    if (m0[k] >= 0) {
      atomicMax(&gt1[m0[k]], __float_as_int((float)(i0 + k)));    // [b][1][m0]
    }
  }
}

// ---------------------------------------------------------------------------
// Host launcher. Stream order gives fill -> scatter -> gather/scatter.
// Deterministic: atomics are max-reductions, fills are total rewrites.
// ---------------------------------------------------------------------------
extern "C" void kernel_launch(void* const* d_in, const int* in_sizes, int n_in,
                              void* d_out, int out_size, void* d_ws, size_t ws_size,
                              hipStream_t stream) {
  const int*   corr   = (const int*)d_in[0];   // facets_corr      [B,F]
  const int*   idx0   = (const int*)d_in[1];   // img0_pooled_idx  [B,N]
  const int*   idx1   = (const int*)d_in[2];   // img1_pooled_idx  [B,N]
  const float* scores = (const float*)d_in[3]; // keypointScores1  [B,N]
  float*       out    = (float*)d_out;         // [B,3,N]
  int*         pos    = (int*)d_ws;

  dim3 blk(BLK);
  int gx4 = (kN4 + BLK - 1) / BLK;             // 196 blocks over N/4

  size_t needed = (size_t)kB * kF * sizeof(int);   // 201 MB batched pos
  if (ws_size >= needed) {
    // Fully batched path: batch in grid.y so dispatch order keeps each
    // batch's 6MB of gather tables temporally hot in L2.
    long long n4 = (long long)kB * kF / 4;
    long long fb = (n4 + BLK - 1) / BLK;
    int fillBlocks = (fb > 8192) ? 8192 : (int)fb;
    fc_fill_neg1_i4<<<fillBlocks, blk, 0, stream>>>((v4i*)pos, n4);
    fc_init_gt1<<<dim3(gx4, kB), blk, 0, stream>>>(out);
    fc_scatter_pos<<<dim3(gx4, kB), blk, 0, stream>>>(idx1, pos);
    fc_main<<<dim3(gx4, kB), blk, 0, stream>>>(corr, idx0, pos, scores, out);
  } else {
    // Fallback: reuse one F-sized pos buffer (3 MB) per batch; stream order
    // serializes the RAW/WAW hazards between batches.
    long long n4 = kF / 4;
    int fillBlocks = (int)((n4 + BLK - 1) / BLK);
    for (int b = 0; b < kB; ++b) {
      fc_fill_neg1_i4<<<fillBlocks, blk, 0, stream>>>((v4i*)pos, n4);
      fc_init_gt1<<<dim3(gx4, 1), blk, 0, stream>>>(out + (long long)b * 3 * kN);
      fc_scatter_pos<<<dim3(gx4, 1), blk, 0, stream>>>(idx1 + (long long)b * kN, pos);
      fc_main<<<dim3(gx4, 1), blk, 0, stream>>>(corr + (long long)b * kF,
                                                idx0 + (long long)b * kN,
                                                pos,
                                                scores + (long long)b * kN,
                                                out + (long long)b * 3 * kN);
    }
  }
}